// HeteroGNN_9706626089208
// MI455X (gfx1250) — compile-verified
//
#include <hip/hip_runtime.h>

#define NNODES 50000
#define FDIM 128
#define OUTF 16

typedef __attribute__((ext_vector_type(2))) float v2f;
typedef __attribute__((ext_vector_type(8))) float v8f;

// ---------------------------------------------------------------- utilities
__global__ void zero_kernel(float* __restrict__ p, long n) {
  long i = (long)blockIdx.x * blockDim.x + threadIdx.x;
  long stride = (long)gridDim.x * blockDim.x;
  for (; i < n; i += stride) p[i] = 0.0f;
}

__global__ void count_kernel(const int* __restrict__ dst,
                             float* __restrict__ cnt, int E) {
  int i = blockIdx.x * blockDim.x + threadIdx.x;
  if (i < E) atomicAdd(&cnt[dst[i]], 1.0f);
}

// One wave per edge: lanes cover the 128-float row with float4 gathers,
// then 4 global f32 atomics per lane into the segment-sum buffer.
__global__ __launch_bounds__(256)
void scatter_kernel(const float* __restrict__ x,
                    const int* __restrict__ src,
                    const int* __restrict__ dst,
                    float* __restrict__ sum, int E) {
  int wave = (int)((blockIdx.x * blockDim.x + threadIdx.x) >> 5);
  int lane = threadIdx.x & 31;
  if (wave >= E) return;
  int s = src[wave];
  int d = dst[wave];
  const float4* xrow = (const float4*)(x + (long)s * FDIM);
  float4 v = xrow[lane];
  float* o = sum + (long)d * FDIM + lane * 4;
  atomicAdd(o + 0, v.x);
  atomicAdd(o + 1, v.y);
  atomicAdd(o + 2, v.z);
  atomicAdd(o + 3, v.w);
}

// ------------------------------------------------- fused SAGE conv (fp32 WMMA)
// out[16 rows x 128 cols per block] =
//   lrelu( (sum/max(cnt,1)) @ Wl + bl + xdst @ Wr )
// Implemented as a K=256 GEMM over the concatenated A = [mean | xdst] tile
// staged in LDS, B = [Wl ; Wr] read through L2 (weights are 128 KB, cached).
__global__ __launch_bounds__(256)
void sage_wmma_kernel(const float* __restrict__ sum,
                      const float* __restrict__ cnt,
                      const float* __restrict__ xdst,
                      const float* __restrict__ Wl,
                      const float* __restrict__ bl,
                      const float* __restrict__ Wr,
                      float* __restrict__ out) {
  __shared__ float As[16][260];  // pad 256 -> 260 floats to break bank conflicts

  const int row0 = blockIdx.x * 16;
  const int tid = threadIdx.x;

  // Cooperative A-tile load: thread `tid` owns column k=tid for all 16 rows.
  for (int r = 0; r < 16; ++r) {
    int row = row0 + r;
    float v;
    if (tid < 128) {
      float c = cnt[row];
      v = sum[(long)row * FDIM + tid] / fmaxf(c, 1.0f);
    } else {
      v = xdst[(long)row * FDIM + (tid - 128)];
    }
    As[r][tid] = v;
  }
  __syncthreads();

  const int wave = tid >> 5;
  const int lane = tid & 31;
  const int half = lane >> 4;   // 0: lanes 0-15, 1: lanes 16-31
  const int m    = lane & 15;   // A row / B,C,D column within tile
  const int col0 = wave * 16;   // this wave's 16-wide output strip
  const int koff = half * 2;    // A: lanes 16-31 hold K+2,K+3; B mirrors it

  v8f acc = {0.f, 0.f, 0.f, 0.f, 0.f, 0.f, 0.f, 0.f};

  for (int k0 = 0; k0 < 256; k0 += 4) {
    v2f a, b;
    a.x = As[m][k0 + koff];
    a.y = As[m][k0 + koff + 1];
    const float* W = (k0 < 128) ? Wl : Wr;
    int kk = (k0 < 128) ? k0 : (k0 - 128);
    b.x = W[(kk + koff) * FDIM + col0 + m];
    b.y = W[(kk + koff + 1) * FDIM + col0 + m];
    acc = __builtin_amdgcn_wmma_f32_16x16x4_f32(false, a, false, b,
                                                (short)0, acc, false, false);
  }

  // Epilogue: bias (lin_l only, per PyG) + leaky_relu(0.01), fp32 out.
  float bias = bl[col0 + m];
#pragma unroll
  for (int r = 0; r < 8; ++r) {
    int mr = r + half * 8;  // C/D layout: VGPR r -> row r (+8 for upper lanes)
    float v = acc[r] + bias;
    v = (v > 0.0f) ? v : 0.01f * v;
    out[(long)(row0 + mr) * FDIM + col0 + m] = v;
  }
}

// ----------------------------------------------------- head GEMM (fp32 WMMA)
// out[N,16] = x[N,128] @ W[128,16] + b ; one wave per 16-row tile.
__global__ __launch_bounds__(256)
void head_wmma_kernel(const float* __restrict__ x,
                      const float* __restrict__ W,
                      const float* __restrict__ b,
                      float* __restrict__ out, int ntiles) {
  const int tid  = threadIdx.x;
  const int wave = tid >> 5;
  const int lane = tid & 31;
  const int tile = blockIdx.x * 8 + wave;
  if (tile >= ntiles) return;  // wave-uniform: EXEC stays all-ones for WMMA

  const int row0 = tile * 16;
  const int half = lane >> 4;
  const int m    = lane & 15;
  const int koff = half * 2;

  v8f acc = {0.f, 0.f, 0.f, 0.f, 0.f, 0.f, 0.f, 0.f};
  for (int k0 = 0; k0 < FDIM; k0 += 4) {
    v2f a, bb;
    a.x = x[(long)(row0 + m) * FDIM + k0 + koff];
    a.y = x[(long)(row0 + m) * FDIM + k0 + koff + 1];
    bb.x = W[(k0 + koff) * OUTF + m];
    bb.y = W[(k0 + koff + 1) * OUTF + m];
    acc = __builtin_amdgcn_wmma_f32_16x16x4_f32(false, a, false, bb,
                                                (short)0, acc, false, false);
  }
  float bias = b[m];
#pragma unroll
  for (int r = 0; r < 8; ++r) {
    int mr = r + half * 8;
    out[(long)(row0 + mr) * OUTF + m] = acc[r] + bias;
  }
}

// ------------------------------------------------------------------- driver
extern "C" void kernel_launch(void* const* d_in, const int* in_sizes, int n_in,
                              void* d_out, int out_size, void* d_ws, size_t ws_size,
                              hipStream_t stream) {
  (void)n_in; (void)out_size; (void)ws_size;

  const float* x_source = (const float*)d_in[0];
  const float* x_target = (const float*)d_in[1];
  const float* Wl0_st = (const float*)d_in[2];
  const float* bl0_st = (const float*)d_in[3];
  const float* Wr0_st = (const float*)d_in[4];
  const float* Wl0_ts = (const float*)d_in[5];
  const float* bl0_ts = (const float*)d_in[6];
  const float* Wr0_ts = (const float*)d_in[7];
  // d_in[8..10] = Wl1_st/bl1_st/Wr1_st: dead branch (xt2 is deleted upstream)
  const float* Wl1_ts = (const float*)d_in[11];
  const float* bl1_ts = (const float*)d_in[12];
  const float* Wr1_ts = (const float*)d_in[13];
  const float* lin_W  = (const float*)d_in[14];
  const float* lin_b  = (const float*)d_in[15];
  const int* ei_st = (const int*)d_in[16];
  const int* ei_ts = (const int*)d_in[17];
  const int E = in_sizes[16] / 2;
  const int Nn = NNODES;

  // Workspace layout (floats): cntA | cntB | bufA | bufB | xt | xs
  float* ws   = (float*)d_ws;
  float* cntA = ws;                         // [N]   in-degree for ei_st
  float* cntB = cntA + Nn;                  // [N]   in-degree for ei_ts (both layers)
  float* bufA = cntB + Nn;                  // [N,F] sum0_t, later sum1_s
  float* bufB = bufA + (long)Nn * FDIM;     // [N,F] sum0_s, later xs2
  float* xt   = bufB + (long)Nn * FDIM;     // [N,F]
  float* xs   = xt   + (long)Nn * FDIM;     // [N,F]

  const int tiles = Nn / 16;                          // 3125
  const int scatter_blocks = (E * 32 + 255) / 256;    // one wave per edge

  // Zero cntA,cntB,bufA,bufB (contiguous region).
  zero_kernel<<<2048, 256, 0, stream>>>(cntA, 2L * Nn + 2L * Nn * FDIM);

  // Degrees.
  count_kernel<<<(E + 255) / 256, 256, 0, stream>>>(ei_st + E, cntA, E);
  count_kernel<<<(E + 255) / 256, 256, 0, stream>>>(ei_ts + E, cntB, E);

  // Layer 0 segment sums.
  scatter_kernel<<<scatter_blocks, 256, 0, stream>>>(x_source, ei_st, ei_st + E, bufA, E);
  scatter_kernel<<<scatter_blocks, 256, 0, stream>>>(x_target, ei_ts, ei_ts + E, bufB, E);

  // Layer 0 fused mean-normalize + dual GEMM + bias + leaky_relu.
  sage_wmma_kernel<<<tiles, 256, 0, stream>>>(bufA, cntA, x_target,
                                              Wl0_st, bl0_st, Wr0_st, xt);
  sage_wmma_kernel<<<tiles, 256, 0, stream>>>(bufB, cntB, x_source,
                                              Wl0_ts, bl0_ts, Wr0_ts, xs);

  // Layer 1 ("ts" branch only): aggregate xt over ei_ts into bufA.
  zero_kernel<<<2048, 256, 0, stream>>>(bufA, (long)Nn * FDIM);
  scatter_kernel<<<scatter_blocks, 256, 0, stream>>>(xt, ei_ts, ei_ts + E, bufA, E);
  sage_wmma_kernel<<<tiles, 256, 0, stream>>>(bufA, cntB, xs,
                                              Wl1_ts, bl1_ts, Wr1_ts, bufB);

  // Head: out = xs2 @ lin_W + lin_b.
  head_wmma_kernel<<<(tiles + 7) / 8, 256, 0, stream>>>(bufB, lin_W, lin_b,
                                                        (float*)d_out, tiles);
}